// FactorizedLayer_45629732553452
// MI455X (gfx1250) — compile-verified
//
#include <hip/hip_runtime.h>

// ---------------------------------------------------------------------------
// FactorizedLayer on MI455X (gfx1250, wave32, WMMA).
//
//   s[b,h,l]   = sum_n x[b,n]   * beta[h,l,n]          (GEMM 4096x8192x512)
//   ssq[b,h,l] = sum_n x[b,n]^2 * beta[h,l,n]^2        (same GEMM, squared ops)
//   out[b,h]   = 0.5*sum_l (s^2 - ssq) + x@W^T + bias
//
// Full-f32 path via V_WMMA_F32_16X16X4_F32 to match the f32 reference exactly
// (the s^2 - ssq subtraction is cancellation-prone, so no bf16 downconvert).
// beta's (H,L,N) layout and W's (H,N) layout are already "B^T row-major",
// i.e. exactly the column-major B operand the WMMA B-fragment wants.
// ---------------------------------------------------------------------------

typedef __attribute__((ext_vector_type(2))) float v2f;
typedef __attribute__((ext_vector_type(8))) float v8f;

#define B_SZ 4096
#define N_SZ 512
#define H_SZ 1024
#define L_SZ 8

static __device__ __forceinline__ v8f wmma_f32(v2f a, v2f b, v8f c) {
    // 8 args: (neg_a, A, neg_b, B, c_mod, C, reuse_a, reuse_b)
    return __builtin_amdgcn_wmma_f32_16x16x4_f32(false, a, false, b,
                                                 (short)0, c, false, false);
}

// ---------------------------------------------------------------------------
// Kernel 1: out[b,h] = x @ W^T + bias  (4096 x 1024 x 512)
// One wave per 16x16 output tile. 8 waves / block.
// ---------------------------------------------------------------------------
__global__ __launch_bounds__(256) void fl_linear_kernel(
        const float* __restrict__ x,
        const float* __restrict__ W,
        const float* __restrict__ bias,
        float* __restrict__ out) {
    const int lane  = threadIdx.x & 31;
    const int wid   = blockIdx.x * 8 + (threadIdx.x >> 5);
    const int colTiles = H_SZ / 16;              // 64
    const int rowBase  = (wid / colTiles) * 16;  // b tile
    const int colBase  = (wid % colTiles) * 16;  // h tile

    const int half  = lane >> 4;   // selects K pair (k = 2*half + j)
    const int idx16 = lane & 15;   // M for A-frag, N for B-frag

    // A-frag: x[rowBase+idx16, k0 + 2*half + {0,1}]   (b64 per lane per step)
    const float* aPtr = x + (size_t)(rowBase + idx16) * N_SZ + 2 * half;
    // B-frag: W[colBase+idx16, k0 + 2*half + {0,1}]   (W is H x N row-major)
    const float* bPtr = W + (size_t)(colBase + idx16) * N_SZ + 2 * half;

    v8f acc = {};
    #pragma unroll 8
    for (int k = 0; k < N_SZ; k += 4) {
        v2f a = *(const v2f*)(aPtr + k);
        v2f b = *(const v2f*)(bPtr + k);
        acc = wmma_f32(a, b, acc);
    }

    const float bv = bias[colBase + idx16];
    #pragma unroll
    for (int r = 0; r < 8; ++r) {
        const int m = rowBase + r + 8 * half;     // C/D layout: M = r + 8*(lane/16)
        out[(size_t)m * H_SZ + colBase + idx16] = acc[r] + bv;
    }
}

// ---------------------------------------------------------------------------
// Kernel 2: FM term. One wave per 16(M) x 32(N) tile of the (B, H*L) GEMM —
// the x A-fragment is reused across two 16-wide column tiles (4 WMMAs/step).
// Squared operands for ssq are produced in-register (packed v_mul), no extra
// memory traffic. Epilogue: 0.5*(s^2 - ssq), reduce over l (8 adjacent
// columns) via shfl_xor 1/2/4 inside aligned 8-lane groups, then += into out
// (stream-ordered after the linear kernel).
// ---------------------------------------------------------------------------
__global__ __launch_bounds__(256) void fl_fm_kernel(
        const float* __restrict__ x,
        const float* __restrict__ beta,
        float* __restrict__ out) {
    const int lane = threadIdx.x & 31;
    const int wid  = blockIdx.x * 8 + (threadIdx.x >> 5);
    const int colTiles = (H_SZ * L_SZ) / 32;     // 256 (32 cols per wave)
    const int rowBase  = (wid / colTiles) * 16;
    const int colBase  = (wid % colTiles) * 32;  // in (h*8 + l) column space

    const int half  = lane >> 4;
    const int idx16 = lane & 15;

    const float* aPtr  = x    + (size_t)(rowBase + idx16) * N_SZ + 2 * half;
    const float* b0Ptr = beta + (size_t)(colBase + idx16) * N_SZ + 2 * half;
    const float* b1Ptr = b0Ptr + (size_t)16 * N_SZ;

    v8f s0 = {}, q0 = {}, s1 = {}, q1 = {};
    #pragma unroll 4
    for (int k = 0; k < N_SZ; k += 4) {
        v2f a  = *(const v2f*)(aPtr  + k);
        v2f b0 = *(const v2f*)(b0Ptr + k);
        v2f b1 = *(const v2f*)(b1Ptr + k);
        v2f a2  = a  * a;
        v2f b0s = b0 * b0;
        v2f b1s = b1 * b1;
        s0 = wmma_f32(a,  b0,  s0);
        q0 = wmma_f32(a2, b0s, q0);
        s1 = wmma_f32(a,  b1,  s1);
        q1 = wmma_f32(a2, b1s, q1);
    }

    #pragma unroll
    for (int r = 0; r < 8; ++r) {
        float v0 = 0.5f * (s0[r] * s0[r] - q0[r]);
        float v1 = 0.5f * (s1[r] * s1[r] - q1[r]);
        // reduce over l = column % 8  (aligned groups of 8 lanes on wave32)
        v0 += __shfl_xor(v0, 1, 32);
        v0 += __shfl_xor(v0, 2, 32);
        v0 += __shfl_xor(v0, 4, 32);
        v1 += __shfl_xor(v1, 1, 32);
        v1 += __shfl_xor(v1, 2, 32);
        v1 += __shfl_xor(v1, 4, 32);
        if ((lane & 7) == 0) {
            const int m      = rowBase + r + 8 * half;
            const int hInTle = (lane >> 3) & 1;          // which 8-col group
            const int h0 = (colBase >> 3) + hInTle;      // first 16-col tile
            const int h1 = ((colBase + 16) >> 3) + hInTle;
            out[(size_t)m * H_SZ + h0] += v0;
            out[(size_t)m * H_SZ + h1] += v1;
        }
    }
}

extern "C" void kernel_launch(void* const* d_in, const int* in_sizes, int n_in,
                              void* d_out, int out_size, void* d_ws, size_t ws_size,
                              hipStream_t stream) {
    (void)in_sizes; (void)n_in; (void)out_size; (void)d_ws; (void)ws_size;

    const float* x    = (const float*)d_in[0];  // (B, N)
    const float* beta = (const float*)d_in[1];  // (H, L, N)
    const float* W    = (const float*)d_in[2];  // (H, N)
    const float* bias = (const float*)d_in[3];  // (H,)
    float* out = (float*)d_out;                 // (B, H)

    // Kernel 1: 256 row-tiles x 64 col-tiles = 16384 waves, 8 waves/block.
    {
        const int nWaves = (B_SZ / 16) * (H_SZ / 16);
        fl_linear_kernel<<<nWaves / 8, 256, 0, stream>>>(x, W, bias, out);
    }
    // Kernel 2: 256 row-tiles x 256 (32-wide) col-tiles = 65536 waves.
    {
        const int nWaves = (B_SZ / 16) * ((H_SZ * L_SZ) / 32);
        fl_fm_kernel<<<nWaves / 8, 256, 0, stream>>>(x, beta, out);
    }
}